// CRF_7155415515267
// MI455X (gfx1250) — compile-verified
//
#include <hip/hip_runtime.h>
#include <hip/hip_bf16.h>

typedef float v2f __attribute__((ext_vector_type(2)));
typedef float v8f __attribute__((ext_vector_type(8)));

#define IMPOSSIBLE -10000.0f
#define B_  256
#define L_  512
#define D_  768
#define C_  19
#define CP  32            // C padded to 32 (two 16-wide WMMA N-tiles)
#define KSTEPS (D_ / 4)   // 192 k-steps of 4 for f32 WMMA

// ---- workspace layout (bytes) ----
#define EMIS_BYTES ((size_t)B_ * L_ * CP * 4)       // 16,777,216
#define WT_BYTES   ((size_t)KSTEPS * 2 * 64 * 4)    // 98,304
#define BP_BYTES   ((size_t)B_ * L_ * CP)           // 4,194,304 (uchar backpointers)

// Pack W_fc (C x D, row-major) into the exact WMMA B-operand stream, with each
// lane's two B-VGPR values ADJACENT so the GEMM loads them as one b64:
//   slot = (kk*2 + g)*64 + lane*2 + v
//   K = 4*kk + v + 2*(lane>>4),  N = 16*g + (lane&15)   (N >= 19 zero-padded)
__global__ void crf_pack_w(const float* __restrict__ W, float* __restrict__ Wt) {
    int s = blockIdx.x * blockDim.x + threadIdx.x;
    if (s >= KSTEPS * 2 * 64) return;
    int kkg  = s >> 6;
    int e    = s & 63;
    int lane = e >> 1;
    int v    = e & 1;
    int kk   = kkg >> 1;
    int g    = kkg & 1;
    int K    = kk * 4 + v + 2 * (lane >> 4);
    int n    = g * 16 + (lane & 15);
    Wt[s] = (n < C_) ? W[n * D_ + K] : 0.0f;
}

// Emission GEMM: emis[row, 0..31] = feats[row, :] @ W^T + b   (row = b*L + l)
// One wave per 16-row tile; v_wmma_f32_16x16x4_f32, two N-groups.
__global__ void __launch_bounds__(128) crf_emis_wmma(
    const float* __restrict__ feats, const float* __restrict__ Wt,
    const float* __restrict__ bias,  float* __restrict__ emis) {
    const int lane = threadIdx.x & 31;
    const int wave = threadIdx.x >> 5;
    const int tile = blockIdx.x * 4 + wave;        // 8192 row tiles
    const int row0 = tile * 16;
    const int half = lane >> 4;                    // K half-pair select
    const int l15  = lane & 15;                    // M (for A) / N (for B,C,D)

    // A operand: lane holds feats[row0+l15, k + 2*half .. +1]  (v2f, 8B aligned)
    const float* A  = feats + (size_t)(row0 + l15) * D_ + 2 * half;
    const float* Bw = Wt + 2 * lane;               // per-lane b64 base

    v8f acc0 = {0.f, 0.f, 0.f, 0.f, 0.f, 0.f, 0.f, 0.f};
    v8f acc1 = {0.f, 0.f, 0.f, 0.f, 0.f, 0.f, 0.f, 0.f};

    #pragma unroll 8
    for (int kk = 0; kk < KSTEPS; ++kk) {
        v2f a  = __builtin_nontemporal_load((const v2f*)(A + kk * 4));
        v2f b0 = *(const v2f*)(Bw + kk * 128);        // N-group 0
        v2f b1 = *(const v2f*)(Bw + kk * 128 + 64);   // N-group 1
        // stream-ahead hint for the A row (lowers to global_prefetch_b8)
        __builtin_prefetch(A + kk * 4 + 256, 0, 0);
        acc0 = __builtin_amdgcn_wmma_f32_16x16x4_f32(
            false, a, false, b0, (short)0, acc0, false, false);
        acc1 = __builtin_amdgcn_wmma_f32_16x16x4_f32(
            false, a, false, b1, (short)0, acc1, false, false);
    }

    float b0v = bias[l15];                       // l15 < 16 < C_ always
    int   n1  = 16 + l15;
    float b1v = (n1 < C_) ? bias[n1] : 0.0f;

    // C/D layout: VGPR r, lanes 0-15 -> M=r, lanes 16-31 -> M=8+r; N = l15 (+16)
    #pragma unroll
    for (int r = 0; r < 8; ++r) {
        int rowm = row0 + r + 8 * half;
        emis[(size_t)rowm * CP + l15]      = acc0[r] + b0v;
        emis[(size_t)rowm * CP + 16 + l15] = acc1[r] + b1v;
    }
}

// Viterbi forward scan + backtrace. One wave32 per batch; lane i owns class i.
__global__ void __launch_bounds__(32) crf_viterbi(
    const float* __restrict__ emis, const float* __restrict__ masks,
    const float* __restrict__ trans, unsigned char* __restrict__ bp,
    float* __restrict__ best_score, int* __restrict__ best_paths) {
    const int b = blockIdx.x;
    const int i = threadIdx.x;          // class (lanes 19..31 are padding)
    const int ic = (i < C_) ? i : 0;

    float tr[C_];
    #pragma unroll
    for (int j = 0; j < C_; ++j) tr[j] = trans[ic * C_ + j];
    const float tstop = trans[(C_ - 1) * C_ + ic];   // stop_idx row

    float score = (i == C_ - 2) ? 0.0f : IMPOSSIBLE; // start_idx = C-2

    for (int t = 0; t < L_; ++t) {
        float e = emis[((size_t)(b * L_ + t)) * CP + i];   // padded, in-bounds
        float m = masks[b * L_ + t];
        float bestv = -3.0e38f;
        int   bestj = 0;
        #pragma unroll
        for (int j = 0; j < C_; ++j) {
            float sj = __shfl(score, j, 32);
            float v  = sj + tr[j];
            if (v > bestv) { bestv = v; bestj = j; }   // strict > : first-idx tie-break
        }
        float acc = bestv + e;
        score = (m > 0.0f) ? acc : score;
        bp[((size_t)(b * L_ + t)) * CP + i] = (unsigned char)bestj;
    }

    score += tstop;
    float bs = -3.0e38f;
    int   bt = 0;
    #pragma unroll
    for (int j = 0; j < C_; ++j) {
        float v = __shfl(score, j, 32);
        if (v > bs) { bs = v; bt = j; }
    }

    if (i == 0) {
        best_score[b] = bs;
        int cur = bt;
        for (int t = L_ - 1; t >= 0; --t) {
            float m = masks[b * L_ + t];
            bool valid = m > 0.0f;
            best_paths[b * L_ + t] = valid ? cur : -1;
            if (valid) cur = (int)bp[((size_t)(b * L_ + t)) * CP + cur];
        }
    }
}

extern "C" void kernel_launch(void* const* d_in, const int* in_sizes, int n_in,
                              void* d_out, int out_size, void* d_ws, size_t ws_size,
                              hipStream_t stream) {
    const float* feats = (const float*)d_in[0];   // (B, L, D) f32
    const float* masks = (const float*)d_in[1];   // (B, L)    f32
    const float* W     = (const float*)d_in[2];   // (C, D)    f32
    const float* bfc   = (const float*)d_in[3];   // (C,)      f32
    const float* trans = (const float*)d_in[4];   // (C, C)    f32

    float* out_score = (float*)d_out;             // (B,) f32
    int*   out_paths = (int*)d_out + B_;          // (B, L) int32 bits

    char* ws = (char*)d_ws;
    float*         emis = (float*)(ws);
    float*         Wt   = (float*)(ws + EMIS_BYTES);
    unsigned char* bp   = (unsigned char*)(ws + EMIS_BYTES + WT_BYTES);

    crf_pack_w<<<(KSTEPS * 2 * 64 + 255) / 256, 256, 0, stream>>>(W, Wt);
    crf_emis_wmma<<<(B_ * L_ / 16) / 4, 128, 0, stream>>>(feats, Wt, bfc, emis);
    crf_viterbi<<<B_, 32, 0, stream>>>(emis, masks, trans, bp, out_score, out_paths);
}